// VOLoss_58944131170642
// MI455X (gfx1250) — compile-verified
//
#include <hip/hip_runtime.h>

#define BB 16
#define NN 1024
#define MM 1024
#define EPSF 1e-8f

typedef __attribute__((ext_vector_type(2))) float v2f;
typedef __attribute__((ext_vector_type(8))) float v8f;

// ---- workspace layout (float offsets) ----
#define OFF_THR   0                       // [16] umax bits during pass1, thr float after prep
#define OFF_ROT   16                      // [16] per-b rotation loss
#define OFF_TRANS 32                      // [16] per-b translation loss
#define OFF_W     48                      // [16] sum of weights
#define OFF_CORR  64                      // [16] sum w*corr_err
#define OFF_EPI   80                      // [16] sum w*epi_err
#define OFF_CNT   96                      // [16] mask count (uint)
// WMMA operand arrays, pre-swizzled into the 16x4 A / 4x16 B lane layout:
// index = ((b*2 + half)*DIM + idx), element = float2 (the lane's 2 VGPRs)
#define OFF_AC    128                     // corr A: half0=(-2T0,-2T1), half1=(S1,1)
#define OFF_AE    (OFF_AC + BB*2*NN*2)    // epi  A: half0=(kp1x,kp1y), half1=(1,0)
#define OFF_BC    (OFF_AE + BB*2*NN*2)    // corr B: half0=(kp2x,kp2y), half1=(1,S2)
#define OFF_BE    (OFF_BC + BB*2*MM*2)    // epi  B: half0=(A0,A1),     half1=(A2,0)

__global__ void k_init(float* ws) {
    if (threadIdx.x < 128) ws[threadIdx.x] = 0.0f;
}

// per-batch max of matches (positive floats -> uint-bit atomicMax is order-preserving)
__global__ void __launch_bounds__(256) k_max(const float* __restrict__ matches, float* ws) {
    const int CH = (NN * MM) / 64;                 // 16384 elems per block
    const int b = blockIdx.x >> 6;
    const int chunk = blockIdx.x & 63;
    const float4* p = (const float4*)(matches + (size_t)b * NN * MM + (size_t)chunk * CH);
    float m = 0.0f;
    for (int i = threadIdx.x; i < CH / 4; i += 256) {
        float4 v = p[i];
        m = fmaxf(m, fmaxf(fmaxf(v.x, v.y), fmaxf(v.z, v.w)));
    }
    __shared__ float s[256];
    s[threadIdx.x] = m; __syncthreads();
    for (int off = 128; off > 0; off >>= 1) {
        if (threadIdx.x < off) s[threadIdx.x] = fmaxf(s[threadIdx.x], s[threadIdx.x + off]);
        __syncthreads();
    }
    if (threadIdx.x == 0)
        atomicMax((unsigned*)(ws + OFF_THR) + b, __float_as_uint(s[0]));
}

// per-batch: pose loss terms, F matrix, pre-swizzled WMMA operand arrays, thr
__global__ void __launch_bounds__(256) k_prep(const float* __restrict__ Rp, const float* __restrict__ tp,
                                              const float* __restrict__ Rg, const float* __restrict__ tg,
                                              const float* __restrict__ K,  const float* __restrict__ kp1,
                                              const float* __restrict__ kp2, float* ws) {
    const int b = blockIdx.x;
    __shared__ float sF[9];
    __shared__ float sR[9];
    if (threadIdx.x == 0) {
        const float* rp = Rp + b * 9;
        const float* rg = Rg + b * 9;
        float tr = 0.0f;
        for (int i = 0; i < 9; ++i) { tr += rp[i] * rg[i]; sR[i] = rg[i]; }
        float geo = (tr - 1.0f) * 0.5f;
        geo = fminf(fmaxf(geo, -1.0f + 1e-7f), 1.0f - 1e-7f);
        ws[OFF_ROT + b] = 1.0f - geo;
        float t0 = tg[b*3], t1 = tg[b*3+1], t2 = tg[b*3+2];
        float inv = 1.0f / sqrtf(t0*t0 + t1*t1 + t2*t2);
        float n0 = t0*inv, n1 = t1*inv, n2 = t2*inv;
        ws[OFF_TRANS + b] = 1.0f - (tp[b*3]*n0 + tp[b*3+1]*n1 + tp[b*3+2]*n2);
        unsigned um = ((unsigned*)(ws + OFF_THR))[b];
        ws[OFF_THR + b] = 0.1f * __uint_as_float(um);
        // K inverse via adjugate
        const float* k = K + b * 9;
        float a = k[0], bb = k[1], c = k[2], d = k[3], e = k[4], f = k[5], g = k[6], h = k[7], ii = k[8];
        float det = a*(e*ii - f*h) - bb*(d*ii - f*g) + c*(d*h - e*g);
        float id = 1.0f / det;
        float Ki[9] = { (e*ii - f*h)*id, (c*h - bb*ii)*id, (bb*f - c*e)*id,
                        (f*g - d*ii)*id, (a*ii - c*g)*id,  (c*d - a*f)*id,
                        (d*h - e*g)*id,  (bb*g - a*h)*id,  (a*e - bb*d)*id };
        float S[9] = { 0.0f, -n2, n1,  n2, 0.0f, -n0,  -n1, n0, 0.0f };
        float E[9], M1[9];
        for (int i = 0; i < 3; ++i) for (int j = 0; j < 3; ++j) {
            float s = 0.0f; for (int kk = 0; kk < 3; ++kk) s += S[i*3+kk] * rg[kk*3+j];
            E[i*3+j] = s;
        }
        for (int i = 0; i < 3; ++i) for (int j = 0; j < 3; ++j) {     // M1 = Ki^T E
            float s = 0.0f; for (int kk = 0; kk < 3; ++kk) s += Ki[kk*3+i] * E[kk*3+j];
            M1[i*3+j] = s;
        }
        for (int i = 0; i < 3; ++i) for (int j = 0; j < 3; ++j) {     // F = M1 Ki
            float s = 0.0f; for (int kk = 0; kk < 3; ++kk) s += M1[i*3+kk] * Ki[kk*3+j];
            sF[i*3+j] = s;
        }
    }
    __syncthreads();
    for (int n = threadIdx.x; n < NN; n += 256) {
        float x = kp1[(b*NN + n)*2], y = kp1[(b*NN + n)*2 + 1];
        float T0 = sR[0]*x + sR[1]*y + sR[2];
        float T1 = sR[3]*x + sR[4]*y + sR[5];
        // corr A operand: K0,K1 = (-2T0,-2T1); K2,K3 = (|T|^2, 1)
        ws[OFF_AC + ((b*2 + 0)*NN + n)*2 + 0] = -2.0f * T0;
        ws[OFF_AC + ((b*2 + 0)*NN + n)*2 + 1] = -2.0f * T1;
        ws[OFF_AC + ((b*2 + 1)*NN + n)*2 + 0] = T0*T0 + T1*T1;
        ws[OFF_AC + ((b*2 + 1)*NN + n)*2 + 1] = 1.0f;
        // epi A operand: K0,K1 = (kp1x,kp1y); K2,K3 = (1, 0)
        ws[OFF_AE + ((b*2 + 0)*NN + n)*2 + 0] = x;
        ws[OFF_AE + ((b*2 + 0)*NN + n)*2 + 1] = y;
        ws[OFF_AE + ((b*2 + 1)*NN + n)*2 + 0] = 1.0f;
        ws[OFF_AE + ((b*2 + 1)*NN + n)*2 + 1] = 0.0f;
    }
    for (int m = threadIdx.x; m < MM; m += 256) {
        float x = kp2[(b*MM + m)*2], y = kp2[(b*MM + m)*2 + 1];
        float A0 = x*sF[0] + y*sF[3] + sF[6];
        float A1 = x*sF[1] + y*sF[4] + sF[7];
        float A2 = x*sF[2] + y*sF[5] + sF[8];
        // corr B operand: K0,K1 = (kp2x,kp2y); K2,K3 = (1, |kp2|^2)
        ws[OFF_BC + ((b*2 + 0)*MM + m)*2 + 0] = x;
        ws[OFF_BC + ((b*2 + 0)*MM + m)*2 + 1] = y;
        ws[OFF_BC + ((b*2 + 1)*MM + m)*2 + 0] = 1.0f;
        ws[OFF_BC + ((b*2 + 1)*MM + m)*2 + 1] = x*x + y*y;
        // epi B operand: K0,K1 = (A0,A1); K2,K3 = (A2, 0)
        ws[OFF_BE + ((b*2 + 0)*MM + m)*2 + 0] = A0;
        ws[OFF_BE + ((b*2 + 0)*MM + m)*2 + 1] = A1;
        ws[OFF_BE + ((b*2 + 1)*MM + m)*2 + 0] = A2;
        ws[OFF_BE + ((b*2 + 1)*MM + m)*2 + 1] = 0.0f;
    }
}

// main pass: WMMA f32 16x16x4 bilinear forms + masked weighted accumulation.
// Hot loop is fully convergent: EXEC is all-1s at every WMMA by construction.
__global__ void __launch_bounds__(256) k_main(const float* __restrict__ matches, float* ws) {
    const int b     = blockIdx.x >> 6;
    const int ntile = blockIdx.x & 63;
    const int wave  = threadIdx.x >> 5;
    const int lane  = threadIdx.x & 31;
    const int row   = lane & 15;       // A: matrix row / B,D: column
    const int half  = lane >> 4;       // 0: K={0,1}, 1: K={2,3} (A/B); D rows +8

    const float thr = ws[OFF_THR + b];
    const int n_r = ntile * 16 + row;

    // straight-line b64 loads, lane layout already matches the WMMA A operand
    const v2f a_corr = ((const v2f*)(ws + OFF_AC))[(b*2 + half)*NN + n_r];
    const v2f a_epi  = ((const v2f*)(ws + OFF_AE))[(b*2 + half)*NN + n_r];
    const v2f* __restrict__ Bc = (const v2f*)(ws + OFF_BC) + (b*2 + half)*MM;
    const v2f* __restrict__ Be = (const v2f*)(ws + OFF_BE) + (b*2 + half)*MM;

    const size_t row_base = ((size_t)b*NN + (size_t)(ntile*16 + half*8)) * MM + row;

    float w_sum = 0.0f, corr_sum = 0.0f, epi_sum = 0.0f;
    unsigned cnt = 0;

    #pragma unroll 2
    for (int mt = wave; mt < 64; mt += 8) {
        const int m_c = mt * 16 + row;
        const v2f b_corr = Bc[m_c];
        const v2f b_epi  = Be[m_c];

        const size_t base = row_base + (size_t)(mt * 16);
        // prefetch this wave's next m-tile of matches into near caches
        __builtin_prefetch(matches + base + 128, 0, 0);

        v8f c0 = {0.0f, 0.0f, 0.0f, 0.0f, 0.0f, 0.0f, 0.0f, 0.0f};
        // dist2[n,m] = S1[n] + S2[m] - 2*T[n].kp2[m] ; g[n,m] = P1[n].A[m]
        v8f dist2 = __builtin_amdgcn_wmma_f32_16x16x4_f32(false, a_corr, false, b_corr, (short)0, c0, false, false);
        v8f gepi  = __builtin_amdgcn_wmma_f32_16x16x4_f32(false, a_epi,  false, b_epi,  (short)0, c0, false, false);

        // D layout: element (M = r + 8*half, N = lane%16) in VGPR r
        #pragma unroll
        for (int r = 0; r < 8; ++r) {
            float mv = matches[base + (size_t)r * MM];
            bool keep = mv > thr;
            float w = keep ? mv : 0.0f;
            cnt += keep ? 1u : 0u;
            float ce = __builtin_amdgcn_sqrtf(fmaxf(dist2[r], 0.0f));  // raw v_sqrt_f32
            float ee = fabsf(gepi[r]);
            w_sum    += w;
            corr_sum += w * ce;
            epi_sum  += w * ee;
        }
    }

    __shared__ float sred[256];
    float vals[3] = {w_sum, corr_sum, epi_sum};
    const int offs[3] = {OFF_W, OFF_CORR, OFF_EPI};
    for (int q = 0; q < 3; ++q) {
        sred[threadIdx.x] = vals[q]; __syncthreads();
        for (int off = 128; off > 0; off >>= 1) {
            if (threadIdx.x < off) sred[threadIdx.x] += sred[threadIdx.x + off];
            __syncthreads();
        }
        if (threadIdx.x == 0) atomicAdd(ws + offs[q] + b, sred[0]);
        __syncthreads();
    }
    __shared__ unsigned scnt[256];
    scnt[threadIdx.x] = cnt; __syncthreads();
    for (int off = 128; off > 0; off >>= 1) {
        if (threadIdx.x < off) scnt[threadIdx.x] += scnt[threadIdx.x + off];
        __syncthreads();
    }
    if (threadIdx.x == 0) atomicAdd((unsigned*)(ws + OFF_CNT) + b, scnt[0]);
}

__global__ void k_final(const float* __restrict__ ws, float* __restrict__ out) {
    if (threadIdx.x == 0) {
        float rot = 0.0f, tra = 0.0f;
        for (int b = 0; b < BB; ++b) { rot += ws[OFF_ROT + b]; tra += ws[OFF_TRANS + b]; }
        float pose = rot / BB + tra / BB;
        const unsigned* cnt = (const unsigned*)(ws + OFF_CNT);
        int nv = 0; float sm = 0.0f, se = 0.0f;
        for (int b = 0; b < BB; ++b) {
            if (cnt[b] >= 3u) {
                ++nv;
                float inv = 1.0f / (ws[OFF_W + b] + EPSF);
                sm += ws[OFF_CORR + b] * inv;
                se += ws[OFF_EPI + b] * inv;
            }
        }
        float ml = (nv > 0) ? sm / nv : 0.0f;
        float el = (nv > 0) ? se / nv : 0.0f;
        out[0] = 1.0f * pose + 0.5f * ml + 0.1f * el;
    }
}

extern "C" void kernel_launch(void* const* d_in, const int* in_sizes, int n_in,
                              void* d_out, int out_size, void* d_ws, size_t ws_size,
                              hipStream_t stream) {
    (void)in_sizes; (void)n_in; (void)out_size; (void)ws_size;
    const float* Rp      = (const float*)d_in[0];
    const float* tp      = (const float*)d_in[1];
    const float* Rg      = (const float*)d_in[2];
    const float* tg      = (const float*)d_in[3];
    const float* K       = (const float*)d_in[4];
    const float* kp1     = (const float*)d_in[5];
    const float* kp2     = (const float*)d_in[6];
    const float* matches = (const float*)d_in[7];
    float* ws  = (float*)d_ws;
    float* out = (float*)d_out;

    k_init <<<1, 128, 0, stream>>>(ws);
    k_max  <<<BB * 64, 256, 0, stream>>>(matches, ws);
    k_prep <<<BB, 256, 0, stream>>>(Rp, tp, Rg, tg, K, kp1, kp2, ws);
    k_main <<<BB * 64, 256, 0, stream>>>(matches, ws);
    k_final<<<1, 32, 0, stream>>>(ws, out);
}